// GraphSAGEModel_67284957659784
// MI455X (gfx1250) — compile-verified
//
#include <hip/hip_runtime.h>

#define NODES   50000
#define EDGES   1000000
#define DIM     64
#define ODIM    8
#define GRAPHS  64

typedef __attribute__((ext_vector_type(2))) float v2f;
typedef __attribute__((ext_vector_type(8))) float v8f;

// ---------------------------------------------------------------- zero
__global__ void zero_kernel(float* __restrict__ p, long n) {
    long i = (long)blockIdx.x * blockDim.x + threadIdx.x;
    long s = (long)gridDim.x * blockDim.x;
    for (; i < n; i += s) p[i] = 0.0f;
}

// ------------------------------------------------- edge scatter (mean aggr)
// one wave32 per edge; lane l moves features [2l, 2l+1] (coalesced float2)
__global__ void scatter_kernel(const float* __restrict__ x,
                               const int*   __restrict__ edge,   // [2, EDGES]
                               float* __restrict__ msg,
                               float* __restrict__ deg) {        // null => skip
    int wave = (int)((blockIdx.x * blockDim.x + threadIdx.x) >> 5);
    int lane = threadIdx.x & 31;
    if (wave >= EDGES) return;
    int src = edge[wave];
    int dst = edge[EDGES + wave];
    float2 v = ((const float2*)(x + (long)src * DIM))[lane];
    float* m = msg + (long)dst * DIM + lane * 2;
    atomicAdd(m,     v.x);
    atomicAdd(m + 1, v.y);
    if (deg != nullptr && lane == 0) atomicAdd(deg + dst, 1.0f);
}

// --------------------------------------- fused SAGE layer, FP32 WMMA 16x16x4
// out[16,64] = relu( mean[16,64] @ Wl + b + x[16,64] @ Wr )
// block = 128 threads = 4 waves; wave w computes N-columns [16w, 16w+16).
// Weights staged in LDS pair-interleaved so every WMMA operand is one
// 8B-aligned ds_load_b64 directly into the even-aligned VGPR pair.
__global__ __launch_bounds__(128)
void sage_wmma_kernel(const float* __restrict__ xin,
                      const float* __restrict__ msg,
                      const float* __restrict__ deg,
                      const float* __restrict__ Wl,   // [64,64] row-major (k,n)
                      const float* __restrict__ bias, // [64]
                      const float* __restrict__ Wr,   // [64,64]
                      float* __restrict__ out) {
    __shared__ float xs[16][DIM + 4];     // stride 68 (even) -> v2f aligned
    __shared__ float ms[16][DIM + 4];
    __shared__ v2f   wlp[DIM / 2][DIM];   // wlp[kp][n] = {Wl[2kp][n], Wl[2kp+1][n]}
    __shared__ v2f   wrp[DIM / 2][DIM];
    const int rowBase = blockIdx.x * 16;
    const int tid = threadIdx.x;

    // stage the 16x64 x-tile and mean-tile into LDS (8 elems / thread)
    #pragma unroll
    for (int i = tid; i < 16 * DIM; i += 128) {
        int r = i >> 6, c = i & 63;
        long node = rowBase + r;
        xs[r][c] = xin[node * DIM + c];
        float d = deg[node];
        d = d > 1.0f ? d : 1.0f;
        ms[r][c] = msg[node * DIM + c] / d;
    }
    // stage weights pair-interleaved (coalesced over n)
    #pragma unroll
    for (int i = tid; i < (DIM / 2) * DIM; i += 128) {
        int kp = i >> 6, c = i & 63;
        v2f wl, wr;
        wl.x = Wl[(2 * kp)     * DIM + c];
        wl.y = Wl[(2 * kp + 1) * DIM + c];
        wr.x = Wr[(2 * kp)     * DIM + c];
        wr.y = Wr[(2 * kp + 1) * DIM + c];
        wlp[kp][c] = wl;
        wrp[kp][c] = wr;
    }
    __syncthreads();

    const int wave = tid >> 5;          // N-tile id 0..3
    const int lane = tid & 31;
    const int colBase = wave * 16;
    const int n  = lane & 15;           // A-row / B-col / C-col within tile
    const int hi = lane >> 4;           // lanes 16..31 hold the K+2/K+3 half

    v8f acc = {};
    // acc += mean @ Wl   (K = 64, 16 steps of K=4)
    #pragma unroll
    for (int k = 0; k < 16; ++k) {
        v2f a = *(const v2f*)&ms[n][4 * k + 2 * hi];
        v2f b = wlp[2 * k + hi][colBase + n];
        acc = __builtin_amdgcn_wmma_f32_16x16x4_f32(
                  false, a, false, b, (short)0, acc, false, false);
    }
    // acc += x @ Wr
    #pragma unroll
    for (int k = 0; k < 16; ++k) {
        v2f a = *(const v2f*)&xs[n][4 * k + 2 * hi];
        v2f b = wrp[2 * k + hi][colBase + n];
        acc = __builtin_amdgcn_wmma_f32_16x16x4_f32(
                  false, a, false, b, (short)0, acc, false, false);
    }

    float bv = bias[colBase + n];
    // C/D layout: VGPR r -> M = r + 8*hi, N = colBase + n
    #pragma unroll
    for (int r = 0; r < 8; ++r) {
        int m = r + (hi ? 8 : 0);
        float v = acc[r] + bv;
        v = v > 0.0f ? v : 0.0f;
        out[(long)(rowBase + m) * DIM + colBase + n] = v;
    }
}

// ---------------------------------------------------------------- pooling
__global__ void pool_kernel(const float* __restrict__ h,
                            const int*   __restrict__ batch,
                            float* __restrict__ psum,
                            float* __restrict__ cnt) {
    int node = (int)((blockIdx.x * blockDim.x + threadIdx.x) >> 5);
    int lane = threadIdx.x & 31;
    if (node >= NODES) return;
    int g = batch[node];
    float2 v = ((const float2*)(h + (long)node * DIM))[lane];
    float* p = psum + (long)g * DIM + lane * 2;
    atomicAdd(p,     v.x);
    atomicAdd(p + 1, v.y);
    if (lane == 0) atomicAdd(cnt + g, 1.0f);
}

// ---------------------------------------------------------------- final fc
__global__ void final_kernel(const float* __restrict__ psum,
                             const float* __restrict__ cnt,
                             const float* __restrict__ Wfc,  // [64,8]
                             const float* __restrict__ bfc,  // [8]
                             float* __restrict__ outp) {
    int i = blockIdx.x * blockDim.x + threadIdx.x;
    if (i >= GRAPHS * ODIM) return;
    int g = i / ODIM, o = i % ODIM;
    float c = cnt[g]; c = c > 1.0f ? c : 1.0f;
    float s = bfc[o];
    #pragma unroll 8
    for (int k = 0; k < DIM; ++k)
        s += (psum[g * DIM + k] / c) * Wfc[k * ODIM + o];
    outp[i] = s;
}

// ---------------------------------------------------------------- launch
extern "C" void kernel_launch(void* const* d_in, const int* in_sizes, int n_in,
                              void* d_out, int out_size, void* d_ws, size_t ws_size,
                              hipStream_t stream) {
    const float* x    = (const float*)d_in[0];
    const int*   edge = (const int*)  d_in[1];   // [2, EDGES]
    const int*   batch= (const int*)  d_in[2];
    const float* W1l  = (const float*)d_in[3];
    const float* b1   = (const float*)d_in[4];
    const float* W1r  = (const float*)d_in[5];
    const float* W2l  = (const float*)d_in[6];
    const float* b2   = (const float*)d_in[7];
    const float* W2r  = (const float*)d_in[8];
    const float* Wfc  = (const float*)d_in[9];
    const float* bfc  = (const float*)d_in[10];
    float* outp = (float*)d_out;

    float* ws   = (float*)d_ws;
    float* msg  = ws;                      // NODES*DIM
    float* deg  = msg  + (long)NODES*DIM;  // NODES
    float* h1   = deg  + NODES;            // NODES*DIM
    float* h2   = h1   + (long)NODES*DIM;  // NODES*DIM
    float* psum = h2   + (long)NODES*DIM;  // GRAPHS*DIM
    float* cnt  = psum + GRAPHS*DIM;       // GRAPHS

    const long msgN = (long)NODES * DIM;
    const int edgeBlocks = (EDGES * 32 + 255) / 256;   // wave per edge
    const int tiles = NODES / 16;                      // 3125, exact

    // layer 1
    zero_kernel<<<2048, 256, 0, stream>>>(msg, msgN + NODES);          // msg+deg
    zero_kernel<<<32, 256, 0, stream>>>(psum, (long)GRAPHS*DIM + GRAPHS);
    scatter_kernel<<<edgeBlocks, 256, 0, stream>>>(x, edge, msg, deg);
    sage_wmma_kernel<<<tiles, 128, 0, stream>>>(x, msg, deg, W1l, b1, W1r, h1);

    // layer 2 (reuse msg, deg unchanged)
    zero_kernel<<<2048, 256, 0, stream>>>(msg, msgN);
    scatter_kernel<<<edgeBlocks, 256, 0, stream>>>(h1, edge, msg, nullptr);
    sage_wmma_kernel<<<tiles, 128, 0, stream>>>(h1, msg, deg, W2l, b2, W2r, h2);

    // pool + fc
    pool_kernel<<<(NODES * 32 + 255) / 256, 256, 0, stream>>>(h2, batch, psum, cnt);
    final_kernel<<<(GRAPHS * ODIM + 255) / 256, 256, 0, stream>>>(psum, cnt, Wfc, bfc, outp);
}